// SSIMLoss_15461882265713
// MI455X (gfx1250) — compile-verified
//
#include <hip/hip_runtime.h>
#include <hip/hip_bf16.h>

// ---------------------------------------------------------------------------
// SSIM loss, bandwidth-bound stencil. CDNA5 path: async global->LDS staging
// (GLOBAL_LOAD_ASYNC_TO_LDS_B128/B32, ASYNCcnt) + LDS separable 3x3 sums.
// ---------------------------------------------------------------------------

#define IMG 512
#define TILE_W 64
#define TILE_H 16
#define HALO_W (TILE_W + 2)            // 66
#define HALO_H (TILE_H + 2)            // 18
// LDS row layout: [pad x3][left halo @3][interior @4..67][right halo @68][pad]
// Interior byte offset = (r*72 + 4)*4, divisible by 16 -> b128-legal.
#define LDS_STRIDE 72
#define LDS_COL0 3                     // LDS column of halo col 0
#define NTHREADS 256

#define SSIM_C1 0.0001f                // 0.01^2
#define SSIM_C2 0.0009f                // 0.03^2

#if defined(__HIP_DEVICE_COMPILE__) && \
    __has_builtin(__builtin_amdgcn_global_load_async_to_lds_b32)
#define USE_ASYNC_LDS 1
#else
#define USE_ASYNC_LDS 0
#endif

#if USE_ASYNC_LDS
// Builtin signatures (from clang diagnostics):
//   b32 : (AS1 int*,  AS3 int*,  imm offset, imm cpol)
//   b128: (AS1 v4i*,  AS3 v4i*,  imm offset, imm cpol)
// Flat->AS1 is value preserving (global aperture is identity).
// Flat->AS3: low 32 bits of a flat LDS address ARE the LDS offset
// (CDNA5 ISA aperture table).
typedef int v4i __attribute__((vector_size(16)));
typedef __attribute__((address_space(1))) int* as1_int_ptr;
typedef __attribute__((address_space(3))) int* as3_int_ptr;
typedef __attribute__((address_space(1))) v4i* as1_v4i_ptr;
typedef __attribute__((address_space(3))) v4i* as3_v4i_ptr;
#define TO_GLOBAL(p)     ((as1_int_ptr)(unsigned long long)(p))
#define TO_LDS(p)        ((as3_int_ptr)(unsigned int)(unsigned long long)(p))
#define TO_GLOBAL_V4(p)  ((as1_v4i_ptr)(unsigned long long)(p))
#define TO_LDS_V4(p)     ((as3_v4i_ptr)(unsigned int)(unsigned long long)(p))

__device__ __forceinline__ void async_cp_b32(const float* g, float* l) {
    __builtin_amdgcn_global_load_async_to_lds_b32(TO_GLOBAL(g), TO_LDS(l), 0, 0);
}
__device__ __forceinline__ void async_cp_b128(const float* g, float* l) {
#if __has_builtin(__builtin_amdgcn_global_load_async_to_lds_b128)
    __builtin_amdgcn_global_load_async_to_lds_b128(TO_GLOBAL_V4(g), TO_LDS_V4(l), 0, 0);
#else
    __builtin_amdgcn_global_load_async_to_lds_b32(TO_GLOBAL(g + 0), TO_LDS(l + 0), 0, 0);
    __builtin_amdgcn_global_load_async_to_lds_b32(TO_GLOBAL(g + 1), TO_LDS(l + 1), 0, 0);
    __builtin_amdgcn_global_load_async_to_lds_b32(TO_GLOBAL(g + 2), TO_LDS(l + 2), 0, 0);
    __builtin_amdgcn_global_load_async_to_lds_b32(TO_GLOBAL(g + 3), TO_LDS(l + 3), 0, 0);
#endif
}
__device__ __forceinline__ void wait_async0() {
#if __has_builtin(__builtin_amdgcn_s_wait_asynccnt)
    __builtin_amdgcn_s_wait_asynccnt(0);
#else
    asm volatile("s_wait_asynccnt 0x0" ::: "memory");
#endif
}
#endif  // USE_ASYNC_LDS

__device__ __forceinline__ int reflect_idx(int i) {
    // jnp.pad mode='reflect', pad=1:  -1 -> 1, 512 -> 510
    if (i < 0)    return -i;
    if (i >= IMG) return 2 * IMG - 2 - i;
    return i;
}

__global__ __launch_bounds__(NTHREADS)
void ssim_tile_kernel(const float* __restrict__ x,
                      const float* __restrict__ y,
                      float* __restrict__ partial) {
    __shared__ float sX[HALO_H * LDS_STRIDE];
    __shared__ float sY[HALO_H * LDS_STRIDE];
    __shared__ float wsum[NTHREADS / 32];

    const int tid = threadIdx.x;
    const int w0  = blockIdx.x * TILE_W;
    const int h0  = blockIdx.y * TILE_H;
    const int img = blockIdx.z;

    const float* __restrict__ xi = x + (size_t)img * IMG * IMG;
    const float* __restrict__ yi = y + (size_t)img * IMG * IMG;

    // ---- Stage (TILE + halo) of x and y into LDS --------------------------
    // Per halo row: 16 x b128 interior chunks + left/right b32 halo loads.
    // Task layout: task = r*18 + t ; t in [0,15]=interior chunk, 16/17=halos.
#define ROW_TASKS 18
#define NTASKS (HALO_H * ROW_TASKS)    // 324
    for (int task = tid; task < NTASKS; task += NTHREADS) {
        const int r = task / ROW_TASKS;
        const int t = task - r * ROW_TASKS;
        const int gh = reflect_idx(h0 + r - 1);
        const float* __restrict__ rowx = xi + (size_t)gh * IMG;
        const float* __restrict__ rowy = yi + (size_t)gh * IMG;
        float* ldsx = &sX[r * LDS_STRIDE];
        float* ldsy = &sY[r * LDS_STRIDE];
        if (t < 16) {
            const int c = t * 4;                       // interior cols c..c+3
#if USE_ASYNC_LDS
            async_cp_b128(rowx + w0 + c, &ldsx[LDS_COL0 + 1 + c]);
            async_cp_b128(rowy + w0 + c, &ldsy[LDS_COL0 + 1 + c]);
#else
            for (int q = 0; q < 4; ++q) {
                ldsx[LDS_COL0 + 1 + c + q] = rowx[w0 + c + q];
                ldsy[LDS_COL0 + 1 + c + q] = rowy[w0 + c + q];
            }
#endif
        } else if (t == 16) {                          // left halo (halo col 0)
            const int gw = reflect_idx(w0 - 1);
#if USE_ASYNC_LDS
            async_cp_b32(rowx + gw, &ldsx[LDS_COL0]);
            async_cp_b32(rowy + gw, &ldsy[LDS_COL0]);
#else
            ldsx[LDS_COL0] = rowx[gw];
            ldsy[LDS_COL0] = rowy[gw];
#endif
        } else {                                       // right halo (halo col 65)
            const int gw = reflect_idx(w0 + TILE_W);
#if USE_ASYNC_LDS
            async_cp_b32(rowx + gw, &ldsx[LDS_COL0 + 1 + TILE_W]);
            async_cp_b32(rowy + gw, &ldsy[LDS_COL0 + 1 + TILE_W]);
#else
            ldsx[LDS_COL0 + 1 + TILE_W] = rowx[gw];
            ldsy[LDS_COL0 + 1 + TILE_W] = rowy[gw];
#endif
        }
    }
#if USE_ASYNC_LDS
    wait_async0();
#endif
    __syncthreads();

    // ---- Separable 3x3 sums from LDS: 1 col x 4 rows per thread -----------
    const int lc = tid & (TILE_W - 1);        // 0..63  (halo cols lc..lc+2)
    const int r0 = (tid >> 6) * 4;            // 0,4,8,12 (halo rows r0..r0+5)

    float hx[6], hy[6], hxx[6], hyy[6], hxy[6];
#pragma unroll
    for (int i = 0; i < 6; ++i) {
        const float* px = &sX[(r0 + i) * LDS_STRIDE + LDS_COL0 + lc];
        const float* py = &sY[(r0 + i) * LDS_STRIDE + LDS_COL0 + lc];
        const float x0 = px[0], x1 = px[1], x2 = px[2];
        const float y0 = py[0], y1 = py[1], y2 = py[2];
        hx[i]  = x0 + x1 + x2;
        hy[i]  = y0 + y1 + y2;
        hxx[i] = x0 * x0 + x1 * x1 + x2 * x2;
        hyy[i] = y0 * y0 + y1 * y1 + y2 * y2;
        hxy[i] = x0 * y0 + x1 * y1 + x2 * y2;
    }

    const float inv9 = 1.0f / 9.0f;
    float acc = 0.0f;
#pragma unroll
    for (int k = 0; k < 4; ++k) {
        const float sx  = hx[k]  + hx[k + 1]  + hx[k + 2];
        const float sy  = hy[k]  + hy[k + 1]  + hy[k + 2];
        const float sxx = hxx[k] + hxx[k + 1] + hxx[k + 2];
        const float syy = hyy[k] + hyy[k + 1] + hyy[k + 2];
        const float sxy = hxy[k] + hxy[k + 1] + hxy[k + 2];

        const float mux   = sx * inv9;
        const float muy   = sy * inv9;
        const float sigx  = sxx * inv9 - mux * mux;
        const float sigy  = syy * inv9 - muy * muy;
        const float sigxy = sxy * inv9 - mux * muy;

        const float n = (2.0f * mux * muy + SSIM_C1) * (2.0f * sigxy + SSIM_C2);
        const float d = (mux * mux + muy * muy + SSIM_C1) * (sigx + sigy + SSIM_C2);
        float v = (1.0f - n / d + 1e-20f) * 0.5f;
        v = fminf(fmaxf(v, 0.0f), 1.0f);
        acc += v;
    }

    // ---- Deterministic reduction: wave32 shuffle + cross-wave LDS ---------
#pragma unroll
    for (int off = 16; off > 0; off >>= 1)
        acc += __shfl_down(acc, off, 32);
    if ((tid & 31) == 0) wsum[tid >> 5] = acc;
    __syncthreads();
    if (tid == 0) {
        float s = 0.0f;
#pragma unroll
        for (int w = 0; w < NTHREADS / 32; ++w) s += wsum[w];
        const int block_flat =
            (blockIdx.z * gridDim.y + blockIdx.y) * gridDim.x + blockIdx.x;
        partial[block_flat] = s;
    }
}

__global__ __launch_bounds__(NTHREADS)
void ssim_reduce_kernel(const float* __restrict__ partial, int n,
                        float* __restrict__ out) {
    __shared__ double red[NTHREADS];
    double s = 0.0;
    for (int i = threadIdx.x; i < n; i += NTHREADS) s += (double)partial[i];
    red[threadIdx.x] = s;
    __syncthreads();
#pragma unroll
    for (int st = NTHREADS / 2; st > 0; st >>= 1) {
        if (threadIdx.x < st) red[threadIdx.x] += red[threadIdx.x + st];
        __syncthreads();
    }
    if (threadIdx.x == 0) {
        const double inv_count = 1.0 / (32.0 * 3.0 * 512.0 * 512.0);
        out[0] = (float)(red[0] * inv_count);
    }
}

extern "C" void kernel_launch(void* const* d_in, const int* in_sizes, int n_in,
                              void* d_out, int out_size, void* d_ws, size_t ws_size,
                              hipStream_t stream) {
    (void)in_sizes; (void)n_in; (void)out_size; (void)ws_size;
    const float* x = (const float*)d_in[0];
    const float* y = (const float*)d_in[1];
    float* out     = (float*)d_out;
    float* partial = (float*)d_ws;   // 8*32*96 = 24576 floats = 96 KiB

    const dim3 grid(IMG / TILE_W, IMG / TILE_H, 32 * 3);   // (8, 32, 96)
    const dim3 block(NTHREADS);
    ssim_tile_kernel<<<grid, block, 0, stream>>>(x, y, partial);

    const int nblocks = (IMG / TILE_W) * (IMG / TILE_H) * 32 * 3;
    ssim_reduce_kernel<<<1, block, 0, stream>>>(partial, nblocks, out);
}